// VoxelLayer_87883620811515
// MI455X (gfx1250) — compile-verified
//
#include <hip/hip_runtime.h>
#include <stdint.h>

#define BINS    (1u << 18)      // global bin capacity (voxel ids clamped for safety)
#define SBLK    512             // scan block size
#define NBLK1   (BINS / SBLK)   // 512 scan blocks -> one 512-wide top-level scan
#define LDSBINS 8192            // per-WG LDS histogram bins (8192*16B = 128 KB of 320 KB WGP LDS)

// ---------------- threefry2x32 (bit-exact vs JAX) ----------------
__host__ __device__ __forceinline__ uint32_t tf_rotl(uint32_t x, int d) {
  return (x << d) | (x >> (32 - d));
}

__host__ __device__ inline void tf2x32(uint32_t k0, uint32_t k1, uint32_t c0, uint32_t c1,
                                       uint32_t* o0, uint32_t* o1) {
  uint32_t ks[3] = {k0, k1, 0x1BD11BDAu ^ k0 ^ k1};
  uint32_t x0 = c0 + ks[0], x1 = c1 + ks[1];
  const int rA[4] = {13, 15, 26, 6};
  const int rB[4] = {17, 29, 16, 24};
  for (int i = 0; i < 5; ++i) {
    const int* r = (i & 1) ? rB : rA;
    for (int j = 0; j < 4; ++j) { x0 += x1; x1 = tf_rotl(x1, r[j]); x1 ^= x0; }
    x0 += ks[(i + 1) % 3];
    x1 += ks[(i + 2) % 3] + (uint32_t)(i + 1);
  }
  *o0 = x0; *o1 = x1;
}

// ---------------- float <-> order-preserving uint ----------------
__device__ __forceinline__ unsigned f2ord(float f) {
  unsigned b = __float_as_uint(f);
  return (b & 0x80000000u) ? ~b : (b | 0x80000000u);
}
__device__ __forceinline__ float ord2f(unsigned u) {
  unsigned b = (u & 0x80000000u) ? (u & 0x7FFFFFFFu) : ~u;
  return __uint_as_float(b);
}

// stats layout (u32): [0..2] min (ordered bits), [3..5] max voxel coord, [6] num_vox
__device__ __forceinline__ unsigned live_bins(const unsigned* __restrict__ stats) {
  long long nb = ((long long)stats[3] + 1) * ((long long)stats[4] + 1) * ((long long)stats[5] + 1);
  return (nb > (long long)BINS) ? BINS : (unsigned)nb;
}

__global__ void k_stats_init(unsigned* __restrict__ stats) {
  unsigned i = threadIdx.x;
  if (i < 3u) stats[i] = 0xFFFFFFFFu;
  else if (i < 7u) stats[i] = 0u;
}

__global__ void k_min(const float* __restrict__ pts, int N, unsigned* __restrict__ stats) {
  __shared__ unsigned sm[3][256];
  int t = threadIdx.x;
  unsigned m0 = 0xFFFFFFFFu, m1 = 0xFFFFFFFFu, m2 = 0xFFFFFFFFu;
  for (int i = blockIdx.x * blockDim.x + t; i < N; i += gridDim.x * blockDim.x) {
    __builtin_prefetch(&pts[i + 2048], 0, 1);   // global_prefetch_b8
    unsigned a = f2ord(pts[i]), b = f2ord(pts[N + i]), c = f2ord(pts[2 * N + i]);
    m0 = a < m0 ? a : m0; m1 = b < m1 ? b : m1; m2 = c < m2 ? c : m2;
  }
  sm[0][t] = m0; sm[1][t] = m1; sm[2][t] = m2;
  __syncthreads();
  for (int off = 128; off > 0; off >>= 1) {
    if (t < off) {
      #pragma unroll
      for (int a = 0; a < 3; ++a) { unsigned v = sm[a][t + off]; if (v < sm[a][t]) sm[a][t] = v; }
    }
    __syncthreads();
  }
  if (t == 0) {
    atomicMin(&stats[0], sm[0][0]);
    atomicMin(&stats[1], sm[1][0]);
    atomicMin(&stats[2], sm[2][0]);
  }
}

__global__ void k_dims(const float* __restrict__ pts, int N, float voxel,
                       unsigned* __restrict__ stats) {
  __shared__ unsigned sm[3][256];
  int t = threadIdx.x;
  float mn0 = ord2f(stats[0]), mn1 = ord2f(stats[1]), mn2 = ord2f(stats[2]);
  unsigned d0 = 0u, d1 = 0u, d2 = 0u;
  for (int i = blockIdx.x * blockDim.x + t; i < N; i += gridDim.x * blockDim.x) {
    __builtin_prefetch(&pts[i + 2048], 0, 1);
    unsigned v0 = (unsigned)(int)floorf((pts[i] - mn0) / voxel);
    unsigned v1 = (unsigned)(int)floorf((pts[N + i] - mn1) / voxel);
    unsigned v2 = (unsigned)(int)floorf((pts[2 * N + i] - mn2) / voxel);
    d0 = v0 > d0 ? v0 : d0; d1 = v1 > d1 ? v1 : d1; d2 = v2 > d2 ? v2 : d2;
  }
  sm[0][t] = d0; sm[1][t] = d1; sm[2][t] = d2;
  __syncthreads();
  for (int off = 128; off > 0; off >>= 1) {
    if (t < off) {
      #pragma unroll
      for (int a = 0; a < 3; ++a) { unsigned v = sm[a][t + off]; if (v > sm[a][t]) sm[a][t] = v; }
    }
    __syncthreads();
  }
  if (t == 0) {
    atomicMax(&stats[3], sm[0][0]);
    atomicMax(&stats[4], sm[1][0]);
    atomicMax(&stats[5], sm[2][0]);
  }
}

// Zero only the live bins (bin < numBins); runs after k_dims.
__global__ void k_zero_sel(const unsigned* __restrict__ stats, unsigned* __restrict__ cnt,
                           float* __restrict__ sx, float* __restrict__ sy,
                           float* __restrict__ sz) {
  unsigned nbins = live_bins(stats);
  for (unsigned i = blockIdx.x * blockDim.x + threadIdx.x; i < nbins;
       i += gridDim.x * blockDim.x) {
    cnt[i] = 0u; sx[i] = 0.f; sy[i] = 0.f; sz[i] = 0.f;
  }
}

__global__ void __launch_bounds__(256)
k_accum(const float* __restrict__ pts, int N, float voxel,
        const unsigned* __restrict__ stats, unsigned* __restrict__ cnt,
        float* __restrict__ sx, float* __restrict__ sy, float* __restrict__ sz) {
  __shared__ unsigned lc[LDSBINS];
  __shared__ float lx[LDSBINS], ly[LDSBINS], lz[LDSBINS];
  int t = threadIdx.x;
  float mn0 = ord2f(stats[0]), mn1 = ord2f(stats[1]), mn2 = ord2f(stats[2]);
  int d1 = (int)stats[4] + 1, d2 = (int)stats[5] + 1;
  unsigned nbins = live_bins(stats);
  bool useLds = (nbins <= LDSBINS);
  if (useLds) {
    for (int i = t; i < LDSBINS; i += blockDim.x) { lc[i] = 0u; lx[i] = 0.f; ly[i] = 0.f; lz[i] = 0.f; }
    __syncthreads();
  }
  for (int i = blockIdx.x * blockDim.x + t; i < N; i += gridDim.x * blockDim.x) {
    __builtin_prefetch(&pts[i + 2048], 0, 1);
    float x = pts[i], y = pts[N + i], z = pts[2 * N + i];
    int v0 = (int)floorf((x - mn0) / voxel);
    int v1 = (int)floorf((y - mn1) / voxel);
    int v2 = (int)floorf((z - mn2) / voxel);
    int lin = (v0 * d1 + v1) * d2 + v2;
    if (useLds) {
      lin = lin < 0 ? 0 : (lin >= LDSBINS ? LDSBINS - 1 : lin);
      atomicAdd(&lc[lin], 1u);                    // ds_add_u32
      atomicAdd(&lx[lin], x);                     // ds_add_f32
      atomicAdd(&ly[lin], y);
      atomicAdd(&lz[lin], z);
    } else {
      lin = lin < 0 ? 0 : (lin >= (int)BINS ? (int)BINS - 1 : lin);
      atomicAdd(&cnt[lin], 1u);
      unsafeAtomicAdd(&sx[lin], x);               // global_atomic_add_f32
      unsafeAtomicAdd(&sy[lin], y);
      unsafeAtomicAdd(&sz[lin], z);
    }
  }
  if (useLds) {
    __syncthreads();
    for (int b = t; b < LDSBINS; b += blockDim.x) {
      unsigned c = lc[b];
      if (c) {
        atomicAdd(&cnt[b], c);
        unsafeAtomicAdd(&sx[b], lx[b]);
        unsafeAtomicAdd(&sy[b], ly[b]);
        unsafeAtomicAdd(&sz[b], lz[b]);
      }
    }
  }
}

__global__ void k_scan1(const unsigned* __restrict__ stats, const unsigned* __restrict__ cnt,
                        unsigned* __restrict__ prefix, unsigned* __restrict__ blockSums) {
  __shared__ unsigned s[SBLK];
  int t = threadIdx.x;
  unsigned nbins = live_bins(stats);
  unsigned bin = blockIdx.x * SBLK + t;
  unsigned flag = (bin < nbins && cnt[bin] > 0u) ? 1u : 0u;
  s[t] = flag;
  __syncthreads();
  for (int off = 1; off < SBLK; off <<= 1) {
    unsigned v = (t >= off) ? s[t - off] : 0u;
    __syncthreads();
    s[t] += v;
    __syncthreads();
  }
  if (bin < nbins) prefix[bin] = s[t] - flag;
  if (t == SBLK - 1) blockSums[blockIdx.x] = s[t];
}

__global__ void k_scan2(const unsigned* __restrict__ blockSums, unsigned* __restrict__ blockOff,
                        unsigned* __restrict__ stats) {
  __shared__ unsigned s[SBLK];
  int t = threadIdx.x;
  unsigned v = blockSums[t];
  s[t] = v;
  __syncthreads();
  for (int off = 1; off < SBLK; off <<= 1) {
    unsigned w = (t >= off) ? s[t - off] : 0u;
    __syncthreads();
    s[t] += w;
    __syncthreads();
  }
  blockOff[t] = s[t] - v;
  if (t == SBLK - 1) stats[6] = s[t];   // num_vox
}

__global__ void k_compact(const unsigned* __restrict__ stats, const unsigned* __restrict__ cnt,
                          const unsigned* __restrict__ prefix, const unsigned* __restrict__ blockOff,
                          const float* __restrict__ sx, const float* __restrict__ sy,
                          const float* __restrict__ sz, float* __restrict__ means) {
  unsigned nbins = live_bins(stats);
  unsigned bin = blockIdx.x * blockDim.x + threadIdx.x;
  if (bin >= nbins) return;
  unsigned c = cnt[bin];
  if (c) {
    unsigned seg = prefix[bin] + blockOff[bin / SBLK];
    float fc = (float)c;
    means[3 * seg + 0] = sx[bin] / fc;
    means[3 * seg + 1] = sy[bin] / fc;
    means[3 * seg + 2] = sz[bin] / fc;
  }
}

__global__ void k_sample(const float* __restrict__ means, const unsigned* __restrict__ stats,
                         float* __restrict__ out, int L, uint32_t k0, uint32_t k1) {
  int j = blockIdx.x * blockDim.x + threadIdx.x;
  if (j >= L) return;
  int half = L >> 1;
  uint32_t e = (uint32_t)((j < half) ? j : (j - half));
  uint32_t o0, o1;
  tf2x32(k0, k1, e, (uint32_t)half + e, &o0, &o1);
  uint32_t bits = (j < half) ? o0 : o1;
  float u = __uint_as_float((bits >> 9) | 0x3F800000u) - 1.0f;   // [0,1), JAX-exact
  int nv = (int)stats[6];
  int idx = (int)(u * (float)nv);
  if (idx > nv - 1) idx = nv - 1;
  out[j]         = means[3 * idx + 0];
  out[L + j]     = means[3 * idx + 1];
  out[2 * L + j] = means[3 * idx + 2];
}

extern "C" void kernel_launch(void* const* d_in, const int* in_sizes, int n_in,
                              void* d_out, int out_size, void* d_ws, size_t ws_size,
                              hipStream_t stream) {
  (void)in_sizes; (void)n_in; (void)out_size; (void)ws_size;
  const float* x = (const float*)d_in[0];
  float* out = (float*)d_out;

  const int B = 16, N1 = 262144, L1 = N1 / 2, L2 = N1 / 4;

  // ---------- workspace layout ----------
  uint8_t* ws = (uint8_t*)d_ws;
  unsigned* stats    = (unsigned*)ws;                         // 256 B (7 u32 used)
  unsigned* cnt      = (unsigned*)(ws + 256);                 // BINS u32
  unsigned* prefix   = cnt + BINS;                            // BINS u32
  unsigned* blockSum = prefix + BINS;                         // 512 u32 (pad to 1024)
  unsigned* blockOff = blockSum + 1024;                       // 512 u32 (pad to 1024)
  float*    sx       = (float*)(blockOff + 1024);             // BINS f32
  float*    sy       = sx + BINS;
  float*    sz       = sy + BINS;
  float*    means    = sz + BINS;                             // 3*BINS f32

  // ---------- JAX key derivation on host (threefry, bit-exact) ----------
  // root = key(42) = (0, 42); split 2: counts [0,1,2,3] -> lanes ([0,1],[2,3])
  uint32_t e0l0, e0l1, e1l0, e1l1;
  tf2x32(0u, 42u, 0u, 2u, &e0l0, &e0l1);
  tf2x32(0u, 42u, 1u, 3u, &e1l0, &e1l1);
  uint32_t kA0 = e0l0, kA1 = e1l0;   // k1 (layer 1)
  uint32_t kB0 = e0l1, kB1 = e1l1;   // k2 (layer 2)
  // split each into 16 batch keys: counts iota(32), elements (i, 16+i)
  uint32_t outA[32], outB[32], keys1[16][2], keys2[16][2];
  for (int i = 0; i < 16; ++i) {
    uint32_t l0, l1;
    tf2x32(kA0, kA1, (uint32_t)i, (uint32_t)(16 + i), &l0, &l1);
    outA[i] = l0; outA[16 + i] = l1;
    tf2x32(kB0, kB1, (uint32_t)i, (uint32_t)(16 + i), &l0, &l1);
    outB[i] = l0; outB[16 + i] = l1;
  }
  for (int b = 0; b < 16; ++b) {
    keys1[b][0] = outA[2 * b]; keys1[b][1] = outA[2 * b + 1];
    keys2[b][0] = outB[2 * b]; keys2[b][1] = outB[2 * b + 1];
  }

  float* out1 = out;
  float* out2 = out + (size_t)B * 3 * L1;

  for (int layer = 0; layer < 2; ++layer) {
    const int   N     = layer ? L1 : N1;
    const int   L     = layer ? L2 : L1;
    const float voxel = layer ? 0.1f : 0.05f;
    const float* src  = layer ? out1 : x;
    float*       dst  = layer ? out2 : out1;

    for (int b = 0; b < B; ++b) {
      const float* pts = src + (size_t)b * 3 * N;
      float*       ob  = dst + (size_t)b * 3 * L;
      uint32_t k0 = layer ? keys2[b][0] : keys1[b][0];
      uint32_t k1 = layer ? keys2[b][1] : keys1[b][1];

      k_stats_init<<<1, 32, 0, stream>>>(stats);
      k_min       <<<256, 256, 0, stream>>>(pts, N, stats);
      k_dims      <<<256, 256, 0, stream>>>(pts, N, voxel, stats);
      k_zero_sel  <<<64, 256, 0, stream>>>(stats, cnt, sx, sy, sz);
      k_accum     <<<256, 256, 0, stream>>>(pts, N, voxel, stats, cnt, sx, sy, sz);
      k_scan1     <<<NBLK1, SBLK, 0, stream>>>(stats, cnt, prefix, blockSum);
      k_scan2     <<<1, SBLK, 0, stream>>>(blockSum, blockOff, stats);
      k_compact   <<<BINS / 256, 256, 0, stream>>>(stats, cnt, prefix, blockOff, sx, sy, sz, means);
      k_sample    <<<(L + 255) / 256, 256, 0, stream>>>(means, stats, ob, L, k0, k1);
    }
  }
}